// BiLSTM_CRF_69561290326779
// MI455X (gfx1250) — compile-verified
//
#include <hip/hip_runtime.h>
#include <hip/hip_bf16.h>
#include <cstdint>
#include <cstddef>

#define SLEN 4096
#define EMB 256
#define HDIR 256
#define G4 1024            // 4*HDIR
#define TTAG 10
#define START_TAG 8
#define STOP_TAG 9
#define NEGV (-10000.0f)

typedef float v2f __attribute__((ext_vector_type(2)));
typedef float v8f __attribute__((ext_vector_type(8)));

// ---------------------------------------------------------------------------
// Kernel A: embedding gather + xW precompute GEMM using fp32 WMMA 16x16x4.
// xw[dir][m][n] = sum_k emb[sent[m]][k] * Wih_dir[n][k] + bih_dir[n] + bhh_dir[n]
// One wave per 16x16 output tile; K looped in steps of 4.
// ---------------------------------------------------------------------------
__global__ __launch_bounds__(256) void xw_gemm_kernel(
    const int* __restrict__ sent,
    const float* __restrict__ emb,
    const float* __restrict__ Wih_f, const float* __restrict__ bih_f,
    const float* __restrict__ bhh_f,
    const float* __restrict__ Wih_b, const float* __restrict__ bih_b,
    const float* __restrict__ bhh_b,
    float* __restrict__ xw)
{
  const int wave = (blockIdx.x << 3) + (threadIdx.x >> 5);   // 32768 waves
  const int lane = threadIdx.x & 31;
  const int mt = wave >> 7;          // 0..255  (SLEN/16)
  const int nt = wave & 127;         // 0..127  (2048/16)
  const int lrow = lane & 15;
  const int khi = (lane >> 4) << 1;  // 0 or 2

  const int m = (mt << 4) + lrow;
  const int n = (nt << 4) + lrow;

  const float* arow = emb + (size_t)sent[m] * EMB;
  const float* wrow;
  float bias;
  int dir, nl;
  if (n < G4) { dir = 0; nl = n;
                wrow = Wih_f + (size_t)nl * EMB; bias = bih_f[nl] + bhh_f[nl]; }
  else        { dir = 1; nl = n - G4;
                wrow = Wih_b + (size_t)nl * EMB; bias = bih_b[nl] + bhh_b[nl]; }

  v8f acc = {0.f, 0.f, 0.f, 0.f, 0.f, 0.f, 0.f, 0.f};
#pragma unroll 8
  for (int k0 = 0; k0 < EMB; k0 += 4) {
    v2f a = *(const v2f*)(arow + k0 + khi);
    v2f b = *(const v2f*)(wrow + k0 + khi);
    acc = __builtin_amdgcn_wmma_f32_16x16x4_f32(
        false, a, false, b, (short)0, acc, false, false);
  }

  const int mrow0 = (mt << 4) + ((lane >> 4) << 3);
  float* op = xw + (size_t)dir * SLEN * G4 + (size_t)mrow0 * G4 + nl;
#pragma unroll
  for (int r = 0; r < 8; ++r)
    op[(size_t)r * G4] = acc[r] + bias;
}

// ---------------------------------------------------------------------------
// Kernel B: sequential LSTM scan. grid = 8 blocks (dir = blk>>2, slice j=blk&3),
// 256 threads each. Each block keeps its 256 rows of Whh (256 KB) in LDS,
// laid out [k4][row][4] so ds_load_b128 across lanes is conflict-free.
// Whh and per-step h are staged into LDS with GLOBAL_LOAD_ASYNC_TO_LDS
// (ASYNCcnt + s_wait_asynccnt); h staging uses scope:SCOPE_DEV for coherence
// with the other 3 WGPs' device-scope stores. Cross-WGP sync: one L2 arrival
// counter per step + double-buffered h — all 8 WGs are co-resident.
// ---------------------------------------------------------------------------
__global__ __launch_bounds__(256) void lstm_scan_kernel(
    const float* __restrict__ Whh_f, const float* __restrict__ Whh_b,
    const float* __restrict__ xw,    // [2][SLEN][1024]
    float* __restrict__ hout,        // [2][SLEN][256]
    float* hbuf,                     // [2][2][256], pre-zeroed
    int* ctr)                        // [2][SLEN],   pre-zeroed
{
  extern __shared__ float smem[];
  float* wl   = smem;                   // 64*256*4 floats = 256 KB
  float* hcur = smem + 64 * 256 * 4;    // 256
  float* gact = hcur + 256;             // 256

  const int tid = threadIdx.x;
  const int dir = blockIdx.x >> 2;
  const int j   = blockIdx.x & 3;
  const int g   = tid >> 6;             // gate: 0=i 1=f 2=g 3=o
  const int q   = tid & 63;
  const int n   = g * HDIR + (j << 6) + q;   // Whh row / xw column
  const int hd  = (j << 6) + q;              // owned hidden dim

  const float* Whh = (dir ? Whh_b : Whh_f) + (size_t)n * HDIR;
  const float* xwd = xw + (size_t)dir * SLEN * G4;
  float* hod = hout + (size_t)dir * SLEN * HDIR;
  float* hb  = hbuf + dir * 2 * HDIR;
  int*   cd  = ctr + dir * SLEN;

  // low 32 bits of a generic LDS pointer == workgroup-relative LDS address
  const unsigned wl_lds   = (unsigned)(uintptr_t)wl;
  const unsigned hcur_lds = (unsigned)(uintptr_t)(hcur + tid);

  // Async-stage this block's Whh rows into LDS, layout wl[(k4*256+tid)*4+c].
  // Per-lane LDS dest addresses build the transposed-interleaved layout
  // directly during the global->LDS copy.
  for (int k4 = 0; k4 < 64; ++k4) {
    unsigned dst = wl_lds + ((((unsigned)k4 << 8) + (unsigned)tid) << 4);
    unsigned long long src = (unsigned long long)(uintptr_t)(Whh + (k4 << 2));
    asm volatile("global_load_async_to_lds_b128 %0, %1, off"
                 :: "v"(dst), "v"(src) : "memory");
  }
  asm volatile("s_wait_asynccnt 0x0" ::: "memory");
  __syncthreads();

  float c = 0.f;
  for (int t = 0; t < SLEN; ++t) {
    // stage h_t straight into LDS; DEV scope -> read from L2 (coherent with
    // the other WGPs' agent-scope stores). buf 0 is pre-zeroed for t=0.
    {
      unsigned long long hsrc =
          (unsigned long long)(uintptr_t)(hb + ((t & 1) << 8) + tid);
      asm volatile("global_load_async_to_lds_b32 %0, %1, off scope:SCOPE_DEV"
                   :: "v"(hcur_lds), "v"(hsrc) : "memory");
      asm volatile("s_wait_asynccnt 0x0" ::: "memory");
    }
    __syncthreads();

    const int ts = dir ? (SLEN - 1 - t) : t;
    float acc = xwd[(size_t)ts * G4 + n];
    const float4* wp = (const float4*)wl;
    const float4* hp = (const float4*)hcur;
#pragma unroll 8
    for (int k4 = 0; k4 < 64; ++k4) {
      float4 w  = wp[(k4 << 8) + tid];
      float4 h4 = hp[k4];
      acc += w.x * h4.x + w.y * h4.y + w.z * h4.z + w.w * h4.w;
    }
    float a = (g == 2) ? tanhf(acc) : 1.f / (1.f + __expf(-acc));
    gact[tid] = a;

    // prefetch next step's xW row while we finish this step
    if (t + 1 < SLEN) {
      const int tn = dir ? (SLEN - 2 - t) : (t + 1);
      __builtin_prefetch(xwd + (size_t)tn * G4 + n, 0, 1);
    }
    __syncthreads();

    if (tid < 64) {
      float iv = gact[tid], fv = gact[64 + tid];
      float gv = gact[128 + tid], ov = gact[192 + tid];
      c = fv * c + iv * gv;
      float h = ov * tanhf(c);
      hod[(size_t)ts * HDIR + hd] = h;
      __hip_atomic_store(hb + (((t + 1) & 1) << 8) + hd, h,
                         __ATOMIC_RELAXED, __HIP_MEMORY_SCOPE_AGENT);
    }
    __threadfence();
    __syncthreads();
    if (tid == 0) {
      __hip_atomic_fetch_add(cd + t, 1, __ATOMIC_RELEASE,
                             __HIP_MEMORY_SCOPE_AGENT);
      while (__hip_atomic_load(cd + t, __ATOMIC_ACQUIRE,
                               __HIP_MEMORY_SCOPE_AGENT) < 4) {
        __builtin_amdgcn_s_sleep(1);
      }
    }
    __syncthreads();
  }
}

// ---------------------------------------------------------------------------
// Kernel C: feats[t][j] = b_out[j] + [h_f(t);h_b(t)] . W_out[j]
// ---------------------------------------------------------------------------
__global__ __launch_bounds__(32) void feats_kernel(
    const float* __restrict__ hout, const float* __restrict__ Wout,
    const float* __restrict__ bout, float* __restrict__ feats)
{
  const int t = blockIdx.x;
  const int jt = threadIdx.x;
  if (jt >= TTAG) return;
  const float* hf = hout + (size_t)t * HDIR;
  const float* hb = hout + (size_t)SLEN * HDIR + (size_t)t * HDIR;
  const float* w = Wout + jt * 512;
  float acc = bout[jt];
  for (int k = 0; k < HDIR; k += 4) {
    float4 a = *(const float4*)(hf + k);
    float4 b = *(const float4*)(w + k);
    acc += a.x * b.x + a.y * b.y + a.z * b.z + a.w * b.w;
    float4 a2 = *(const float4*)(hb + k);
    float4 b2 = *(const float4*)(w + HDIR + k);
    acc += a2.x * b2.x + a2.y * b2.y + a2.z * b2.z + a2.w * b2.w;
  }
  feats[t * TTAG + jt] = acc;
}

// ---------------------------------------------------------------------------
// Kernel D: Viterbi, single wave32 (workgroup barriers are S_NOPs).
// Backpointers kept in LDS (u8, 40 KB) so the serial backtrace stays on-chip.
// out[0] = score, out[1..SLEN] = path (as float, per harness output dtype).
// ---------------------------------------------------------------------------
__global__ __launch_bounds__(32) void viterbi_kernel(
    const float* __restrict__ feats, const float* __restrict__ trans,
    float* __restrict__ out)
{
  __shared__ float fv[16];
  __shared__ unsigned char bps[SLEN * TTAG];   // 40 KB
  const int jt = threadIdx.x;
  float tr[TTAG];
  if (jt < TTAG) {
    for (int i = 0; i < TTAG; ++i) tr[i] = trans[jt * TTAG + i];
    fv[jt] = (jt == START_TAG) ? 0.f : NEGV;
  }
  __syncthreads();
  for (int t = 0; t < SLEN; ++t) {
    float best = -3.4e38f; int bi = 0;
    if (jt < TTAG) {
      for (int i = 0; i < TTAG; ++i) {
        float v = fv[i] + tr[i];
        if (v > best) { best = v; bi = i; }   // first-index tie break
      }
      bps[t * TTAG + jt] = (unsigned char)bi;
      best += feats[t * TTAG + jt];
    }
    __syncthreads();
    if (jt < TTAG) fv[jt] = best;
    __syncthreads();
  }
  if (jt == 0) {
    float bestv = -3.4e38f; int tag = 0;
    for (int i = 0; i < TTAG; ++i) {
      float v = fv[i] + trans[STOP_TAG * TTAG + i];
      if (v > bestv) { bestv = v; tag = i; }
    }
    out[0] = bestv;
    out[SLEN] = (float)tag;                 // path[S-1]
    for (int t = SLEN - 1; t >= 1; --t) {
      tag = (int)bps[t * TTAG + tag];
      out[t] = (float)tag;                  // path[t-1]
    }
  }
}

// ---------------------------------------------------------------------------
extern "C" void kernel_launch(void* const* d_in, const int* in_sizes, int n_in,
                              void* d_out, int out_size, void* d_ws,
                              size_t ws_size, hipStream_t stream)
{
  (void)in_sizes; (void)n_in; (void)out_size; (void)ws_size;
  const int*   sent  = (const int*)d_in[0];
  const float* emb   = (const float*)d_in[1];
  const float* Wih_f = (const float*)d_in[2];
  const float* Whh_f = (const float*)d_in[3];
  const float* bih_f = (const float*)d_in[4];
  const float* bhh_f = (const float*)d_in[5];
  const float* Wih_b = (const float*)d_in[6];
  const float* Whh_b = (const float*)d_in[7];
  const float* bih_b = (const float*)d_in[8];
  const float* bhh_b = (const float*)d_in[9];
  const float* Wout  = (const float*)d_in[10];
  const float* bout  = (const float*)d_in[11];
  const float* trans = (const float*)d_in[12];
  float* out = (float*)d_out;

  char* ws = (char*)d_ws;
  float* xw    = (float*)ws;                                   // 32 MB
  float* hout  = (float*)(ws + (size_t)2 * SLEN * G4 * 4);     //  8 MB
  float* feats = (float*)((char*)hout + (size_t)2 * SLEN * HDIR * 4);
  float* hbuf  = (float*)((char*)feats + (size_t)SLEN * TTAG * 4);
  int*   ctr   = (int*)((char*)hbuf + 2 * 2 * HDIR * 4);

  // zero h double-buffer + step arrival counters (re-done every graph replay)
  hipMemsetAsync(hbuf, 0, 2 * 2 * HDIR * 4 + (size_t)2 * SLEN * 4, stream);

  xw_gemm_kernel<<<4096, 256, 0, stream>>>(sent, emb, Wih_f, bih_f, bhh_f,
                                           Wih_b, bih_b, bhh_b, xw);

  const size_t smem = (size_t)(64 * 256 * 4 + 256 + 256) * sizeof(float); // 258 KB
  lstm_scan_kernel<<<8, 256, smem, stream>>>(Whh_f, Whh_b, xw, hout, hbuf, ctr);

  feats_kernel<<<SLEN, 32, 0, stream>>>(hout, Wout, bout, feats);
  viterbi_kernel<<<1, 32, 0, stream>>>(feats, trans, out);
}